// GCRNNCell_20486994002046
// MI455X (gfx1250) — compile-verified
//
#include <hip/hip_runtime.h>
#include <math.h>

// GCRNN cell: B=64, N=4096, F=2, U=16, S=2
#define Bsz 64
#define Nn  4096
#define Ff  2
#define Uu  16
#define Ssup 2

#define KC   16      // K-chunk staged per iteration
#define LDP  20      // padded LDS row stride (floats): bank-conflict-free, 16B-aligned segs

typedef __attribute__((ext_vector_type(2))) float v2f;
typedef __attribute__((ext_vector_type(4))) float v4f;
typedef __attribute__((ext_vector_type(8))) float v8f;
typedef __attribute__((ext_vector_type(4))) int   v4i;

#if defined(__has_builtin)
#  if __has_builtin(__builtin_amdgcn_global_load_async_to_lds_b128)
#    define HAVE_ASYNC_LDS 1
#  endif
#endif
#ifndef HAVE_ASYNC_LDS
#  define HAVE_ASYNC_LDS 0
#endif

#define AS1 __attribute__((address_space(1)))
#define AS3 __attribute__((address_space(3)))

// Out[row, m] = sum_n X[row, n] * Sup[m, n]   (Out = X * Sup^T), K = 4096.
// 256-thread block computes a 128x128 macro tile; A/B K-chunks staged in LDS
// (double-buffered, async-to-LDS when available). Each wave: 32 rows x 64 cols
// = 2 A frags x 4 B frags -> 8 v_wmma_f32_16x16x4_f32 per k-step.
__global__ __launch_bounds__(256)
void diffuse_gemm(const float* __restrict__ X, const float* __restrict__ Sup,
                  float* __restrict__ Out) {
  __shared__ float As[2][128 * LDP];
  __shared__ float Bs[2][128 * LDP];

  const int tid  = threadIdx.x;
  const int lane = tid & 31;
  const int wave = tid >> 5;
  const int l15  = lane & 15;
  const int hi   = lane >> 4;                 // 0 or 1
  const int row0 = blockIdx.y << 7;           // 128-row macro tile
  const int col0 = blockIdx.x << 7;           // 128-col macro tile
  const int wr   = (wave >> 1) << 5;          // wave row offset: 0,32,64,96
  const int wc   = (wave & 1) << 6;           // wave col offset: 0,64

  // Staging: each thread moves one float4 from two rows of A and two of B.
  const int srow = tid >> 2;                  // 0..63
  const int sseg = (tid & 3) << 2;            // 0,4,8,12
  const float* gA0 = X   + (size_t)(row0 + srow) * Nn + sseg;
  const float* gA1 = X   + (size_t)(row0 + srow + 64) * Nn + sseg;
  const float* gB0 = Sup + (size_t)(col0 + srow) * Nn + sseg;
  const float* gB1 = Sup + (size_t)(col0 + srow + 64) * Nn + sseg;
  const int la0 = srow * LDP + sseg;
  const int la1 = (srow + 64) * LDP + sseg;

  auto stage = [&](int k0, int buf) {
#if HAVE_ASYNC_LDS
    __builtin_amdgcn_global_load_async_to_lds_b128(
        (AS1 v4i*)(gA0 + k0), (AS3 v4i*)&As[buf][la0], 0, 0);
    __builtin_amdgcn_global_load_async_to_lds_b128(
        (AS1 v4i*)(gA1 + k0), (AS3 v4i*)&As[buf][la1], 0, 0);
    __builtin_amdgcn_global_load_async_to_lds_b128(
        (AS1 v4i*)(gB0 + k0), (AS3 v4i*)&Bs[buf][la0], 0, 0);
    __builtin_amdgcn_global_load_async_to_lds_b128(
        (AS1 v4i*)(gB1 + k0), (AS3 v4i*)&Bs[buf][la1], 0, 0);
#else
    v4f ra0 = *(const v4f*)(gA0 + k0);
    v4f ra1 = *(const v4f*)(gA1 + k0);
    v4f rb0 = *(const v4f*)(gB0 + k0);
    v4f rb1 = *(const v4f*)(gB1 + k0);
    *(v4f*)&As[buf][la0] = ra0;
    *(v4f*)&As[buf][la1] = ra1;
    *(v4f*)&Bs[buf][la0] = rb0;
    *(v4f*)&Bs[buf][la1] = rb1;
#endif
  };

  auto wait_staged = [&]() {
#if HAVE_ASYNC_LDS
#  if __has_builtin(__builtin_amdgcn_s_wait_asynccnt)
    __builtin_amdgcn_s_wait_asynccnt(0);
#  else
    asm volatile("s_wait_asynccnt 0" ::: "memory");
#  endif
#endif
    __syncthreads();
  };

  v8f acc[2][4] = {};

  stage(0, 0);
  wait_staged();

  const int NCHUNK = Nn / KC;                 // 256
  for (int c = 0; c < NCHUNK; ++c) {
    const int buf = c & 1;
    if (c + 1 < NCHUNK) stage((c + 1) * KC, buf ^ 1);

    const float* as = As[buf];
    const float* bs = Bs[buf];
    #pragma unroll
    for (int kk = 0; kk < KC; kk += 4) {
      const int ko = kk + 2 * hi;
      v2f a0 = *(const v2f*)&as[(wr +  0 + l15) * LDP + ko];
      v2f a1 = *(const v2f*)&as[(wr + 16 + l15) * LDP + ko];
      v2f b0 = *(const v2f*)&bs[(wc +  0 + l15) * LDP + ko];
      v2f b1 = *(const v2f*)&bs[(wc + 16 + l15) * LDP + ko];
      v2f b2 = *(const v2f*)&bs[(wc + 32 + l15) * LDP + ko];
      v2f b3 = *(const v2f*)&bs[(wc + 48 + l15) * LDP + ko];
      acc[0][0] = __builtin_amdgcn_wmma_f32_16x16x4_f32(false, a0, false, b0, (short)0, acc[0][0], false, false);
      acc[0][1] = __builtin_amdgcn_wmma_f32_16x16x4_f32(false, a0, false, b1, (short)0, acc[0][1], false, false);
      acc[0][2] = __builtin_amdgcn_wmma_f32_16x16x4_f32(false, a0, false, b2, (short)0, acc[0][2], false, false);
      acc[0][3] = __builtin_amdgcn_wmma_f32_16x16x4_f32(false, a0, false, b3, (short)0, acc[0][3], false, false);
      acc[1][0] = __builtin_amdgcn_wmma_f32_16x16x4_f32(false, a1, false, b0, (short)0, acc[1][0], false, false);
      acc[1][1] = __builtin_amdgcn_wmma_f32_16x16x4_f32(false, a1, false, b1, (short)0, acc[1][1], false, false);
      acc[1][2] = __builtin_amdgcn_wmma_f32_16x16x4_f32(false, a1, false, b2, (short)0, acc[1][2], false, false);
      acc[1][3] = __builtin_amdgcn_wmma_f32_16x16x4_f32(false, a1, false, b3, (short)0, acc[1][3], false, false);
    }
    wait_staged();
  }

  // C/D layout: VGPR i holds row M = i + 8*hi, column = l15.
  #pragma unroll
  for (int g = 0; g < 2; ++g)
    #pragma unroll
    for (int t = 0; t < 4; ++t)
      #pragma unroll
      for (int i = 0; i < 8; ++i)
        Out[(size_t)(row0 + wr + (g << 4) + i + (hi << 3)) * Nn +
            col0 + wc + (t << 4) + l15] = acc[g][t][i];
}

// One thread per (b, m): r/u gates from hs[36] = [h(4), st(32)], write r*prev and u.
__global__ __launch_bounds__(256)
void gate_kernel(const float* __restrict__ H, const float* __restrict__ ST,
                 const float* __restrict__ state,
                 const float* __restrict__ rk, const float* __restrict__ uk,
                 float* __restrict__ RP, float* __restrict__ Ug) {
  __shared__ float srk[36 * Uu];
  __shared__ float suk[36 * Uu];
  for (int i = threadIdx.x; i < 36 * Uu; i += 256) { srk[i] = rk[i]; suk[i] = uk[i]; }
  __syncthreads();

  const int t = blockIdx.x * 256 + threadIdx.x;  // b*N + m
  const int b = t >> 12;
  const int m = t & (Nn - 1);

  float hs[36];
  #pragma unroll
  for (int s = 0; s < Ssup; ++s)
    #pragma unroll
    for (int c = 0; c < Ff; ++c)
      hs[s * Ff + c] = H[((size_t)s * Bsz * Ff + b * Ff + c) * Nn + m];
  #pragma unroll
  for (int s = 0; s < Ssup; ++s)
    #pragma unroll
    for (int u = 0; u < Uu; ++u)
      hs[Ssup * Ff + s * Uu + u] = ST[((size_t)s * Bsz * Uu + b * Uu + u) * Nn + m];

  float racc[Uu], uacc[Uu];
  #pragma unroll
  for (int u = 0; u < Uu; ++u) { racc[u] = 0.f; uacc[u] = 0.f; }
  #pragma unroll
  for (int j = 0; j < 36; ++j) {
    const float x = hs[j];
    #pragma unroll
    for (int u = 0; u < Uu; ++u) {
      racc[u] = fmaf(x, srk[j * Uu + u], racc[u]);
      uacc[u] = fmaf(x, suk[j * Uu + u], uacc[u]);
    }
  }
  #pragma unroll
  for (int u = 0; u < Uu; ++u) {
    const size_t idx = ((size_t)b * Uu + u) * Nn + m;
    const float r  = 1.f / (1.f + expf(-racc[u]));
    const float ug = 1.f / (1.f + expf(-uacc[u]));
    RP[idx] = r * state[idx];
    Ug[idx] = ug;
  }
}

// One thread per (b, m): c = tanh([h, c_state] @ ck); z = u*prev + (1-u)*c.
// Output layout: out[b*U*N + m*U + u]  (z is (B, N, U) flattened).
__global__ __launch_bounds__(256)
void final_kernel(const float* __restrict__ H, const float* __restrict__ CS,
                  const float* __restrict__ Ug, const float* __restrict__ state,
                  const float* __restrict__ ck, float* __restrict__ out) {
  __shared__ float sck[36 * Uu];
  for (int i = threadIdx.x; i < 36 * Uu; i += 256) sck[i] = ck[i];
  __syncthreads();

  const int t = blockIdx.x * 256 + threadIdx.x;  // b*N + m
  const int b = t >> 12;
  const int m = t & (Nn - 1);

  float hc[36];
  #pragma unroll
  for (int s = 0; s < Ssup; ++s)
    #pragma unroll
    for (int c = 0; c < Ff; ++c)
      hc[s * Ff + c] = H[((size_t)s * Bsz * Ff + b * Ff + c) * Nn + m];
  #pragma unroll
  for (int s = 0; s < Ssup; ++s)
    #pragma unroll
    for (int u = 0; u < Uu; ++u)
      hc[Ssup * Ff + s * Uu + u] = CS[((size_t)s * Bsz * Uu + b * Uu + u) * Nn + m];

  float cacc[Uu];
  #pragma unroll
  for (int u = 0; u < Uu; ++u) cacc[u] = 0.f;
  #pragma unroll
  for (int j = 0; j < 36; ++j) {
    const float x = hc[j];
    #pragma unroll
    for (int u = 0; u < Uu; ++u) cacc[u] = fmaf(x, sck[j * Uu + u], cacc[u]);
  }
  #pragma unroll
  for (int u = 0; u < Uu; ++u) {
    const size_t sidx = ((size_t)b * Uu + u) * Nn + m;
    const float cg = tanhf(cacc[u]);
    const float ug = Ug[sidx];
    const float z  = ug * state[sidx] + (1.f - ug) * cg;
    out[((size_t)b * Nn + m) * Uu + u] = z;
  }
}

extern "C" void kernel_launch(void* const* d_in, const int* in_sizes, int n_in,
                              void* d_out, int out_size, void* d_ws, size_t ws_size,
                              hipStream_t stream) {
  const float* inputs = (const float*)d_in[0];  // (B, F*N) == (B*F, N) rows
  const float* state  = (const float*)d_in[1];  // (B, U*N) == (B*U, N) rows
  const float* sup0   = (const float*)d_in[2];  // (N, N)
  const float* sup1   = (const float*)d_in[3];  // (N, N)
  const float* rk     = (const float*)d_in[4];  // (36, 16)
  const float* uk     = (const float*)d_in[5];  // (36, 16)
  const float* ck     = (const float*)d_in[6];  // (36, 16)
  float* out = (float*)d_out;

  float* ws = (float*)d_ws;
  const size_t HN  = (size_t)Ssup * Bsz * Ff * Nn;  //  1,048,576 floats
  const size_t STN = (size_t)Ssup * Bsz * Uu * Nn;  //  8,388,608 floats
  const size_t RPN = (size_t)Bsz * Uu * Nn;         //  4,194,304 floats
  float* H  = ws;            // (S, B*F, N)
  float* ST = H  + HN;       // (S, B*U, N)
  float* RP = ST + STN;      // (B*U, N)
  float* CS = RP + RPN;      // (S, B*U, N)
  float* Ug = CS + STN;      // (B*U, N)

  const dim3 blk(256);
  const int colB = Nn / 128;                  // 32 col macro-tiles
  // Pass 1: h = diffuse(feats), st = diffuse(prev)
  diffuse_gemm<<<dim3(colB, (Bsz * Ff) / 128), blk, 0, stream>>>(inputs, sup0, H);
  diffuse_gemm<<<dim3(colB, (Bsz * Ff) / 128), blk, 0, stream>>>(inputs, sup1, H + (size_t)Bsz * Ff * Nn);
  diffuse_gemm<<<dim3(colB, (Bsz * Uu) / 128), blk, 0, stream>>>(state, sup0, ST);
  diffuse_gemm<<<dim3(colB, (Bsz * Uu) / 128), blk, 0, stream>>>(state, sup1, ST + (size_t)Bsz * Uu * Nn);
  // Gates r, u; RP = r * prev
  gate_kernel<<<dim3((Bsz * Nn) / 256), blk, 0, stream>>>(H, ST, state, rk, uk, RP, Ug);
  // Pass 2: c_state = diffuse(r * prev)  (supports hit L2 here)
  diffuse_gemm<<<dim3(colB, (Bsz * Uu) / 128), blk, 0, stream>>>(RP, sup0, CS);
  diffuse_gemm<<<dim3(colB, (Bsz * Uu) / 128), blk, 0, stream>>>(RP, sup1, CS + (size_t)Bsz * Uu * Nn);
  // c, z, output
  final_kernel<<<dim3((Bsz * Nn) / 256), blk, 0, stream>>>(H, CS, Ug, state, ck, out);
}